// ResidualVectorQuantize_56848187130001
// MI455X (gfx1250) — compile-verified
//
#include <hip/hip_runtime.h>
#include <math.h>

// ---------------------------------------------------------------------------
// Fused Residual Vector Quantize forward for MI455X (gfx1250, wave32, WMMA).
//
// B=16, D=512, T=4096, N_CB=9, K=1024, CDIM=8.
//
//  * One fused kernel; residual tile (16 tokens x 512 dims) lives in LDS for
//    all 9 codebook steps -> HBM traffic ~290MB instead of ~2.4GB.
//  * argmin of cosine distance == argmax of (raw_enc . normalized_codebook);
//    codebook pre-normalized once into d_ws by a helper kernel.
//  * All three GEMMs use V_WMMA_F32_16X16X4_F32 (fp32 inputs, argmin is
//    precision sensitive).
//  * Per-step staging of cbn[i] / w_out[i] uses the Tensor Data Mover with
//    pad_enable (8 DWORD rows padded to 10) when the builtin is available;
//    plain-load fallback otherwise.
//  * Losses: cl == cbl numerically; reduced wave->block->partials->final
//    fixed-order sum kernel (deterministic, no float atomics).
// ---------------------------------------------------------------------------

typedef float v2f __attribute__((ext_vector_type(2)));
typedef float v4f __attribute__((ext_vector_type(4)));
typedef float v8f __attribute__((ext_vector_type(8)));
typedef unsigned int u32x4 __attribute__((ext_vector_type(4)));
typedef int i32x4 __attribute__((ext_vector_type(4)));
typedef int i32x8 __attribute__((ext_vector_type(8)));

#define NB   16
#define DD   512
#define TT   4096
#define NCB  9
#define KC   1024

// d_out element offsets (fp32), outputs concatenated in return order:
// z_q (16*512*4096), codes (16*9*4096, stored as float(idx)),
// latents (16*72*4096), cl, cbl.
#define ZQ_OFF    0ull
#define CODES_OFF 33554432ull
#define LAT_OFF   34144256ull
#define CL_OFF    38862848ull
#define CBL_OFF   38862849ull

// d_ws layout (floats): normalized codebook, then per-block loss partials.
#define WS_CBN_FLOATS (NCB * KC * 8)   // 73728
#define WS_PART_OFF   WS_CBN_FLOATS
#define NBLOCKS       1024

// LDS layout (floats). Residual stored [tok][dim] padded 512->516 (bank
// spread + keeps float2/float4 fragment loads aligned). w_in staged as 16
// padded rows, rows 8..15 zeroed once -> no mask multiply in the K loop.
#define RES_STR  516
#define RES_PW   (16 * RES_STR)             // per-wave residual tile
#define OFF_RES  0
#define OFF_WIN  (4 * RES_PW)               // w_in[i]: 16 rows x 516 (8 live)
#define OFF_WOUT (OFF_WIN + 16 * 516)       // w_out[i]: 512 rows x 10 (pad 8->10)
#define OFF_CBN  (OFF_WOUT + 512 * 10)      // cbn[i]:   1024 rows x 10
#define OFF_BOUT (OFF_CBN + 1024 * 10)      // b_out[i]: 512
#define OFF_BIN  (OFF_BOUT + 512)           // b_in[i]:  8 (pad 16)
#define OFF_ENC  (OFF_BIN + 16)             // per-wave enc tile: 16 x 10
#define LDS_FLOATS (OFF_ENC + 4 * 160)      // 57808 floats = 231,232 B (< 320KB/WGP)

// D = A(16x4) * B(4x16) + C.  A: lane(M=lane&15) holds K={k0,k0+1},
// k0 = (lane<16 ? 0 : 2).  B: lane(N=lane&15) holds same K pair.
// C/D: vgpr j, lane L -> (M = j + 8*(L>>4), N = L&15).
__device__ __forceinline__ v8f wmma4(v2f a, v2f b, v8f c) {
  return __builtin_amdgcn_wmma_f32_16x16x4_f32(
      /*neg_a=*/false, a, /*neg_b=*/false, b,
      /*c_mod=*/(short)0, c, /*reuse_a=*/false, /*reuse_b=*/false);
}

// --------------------------- TDM staging -----------------------------------
#if defined(__gfx1250__) && __has_builtin(__builtin_amdgcn_tensor_load_to_lds) \
    && __has_builtin(__builtin_amdgcn_s_wait_tensorcnt)
#define HAVE_TDM 1
#else
#define HAVE_TDM 0
#endif

#if HAVE_TDM
// Copy `rows` rows of 8 f32 from global into LDS, padding each row 8->10
// DWORDs (TDM pad: interval code 2 = 8 DWORDs, amount code 1 = 2 DWORDs).
__device__ __forceinline__ void tdm_load_rows8_pad10(const float* gsrc,
                                                     unsigned lds_byte_off,
                                                     int rows) {
  u32x4 g0;
  unsigned long long ga = (unsigned long long)gsrc;
  g0.x = 1u;                                        // count=1, user mode
  g0.y = lds_byte_off;                              // LDS byte address
  g0.z = (unsigned)(ga & 0xFFFFFFFFull);            // global_addr[31:0]
  g0.w = (unsigned)((ga >> 32) & 0x01FFFFFFull) | (2u << 30);  // [56:32]+type=2
  i32x8 g1;
  g1[0] = (2 << 16)        // data_size = 4B
        | (1 << 20)        // pad_enable
        | (2 << 22)        // pad_interval: 8 DWORDs
        | (1 << 25);       // pad_amount: 2 DWORDs
  g1[1] = (8 << 16);       // tensor_dim0 = 8 (bits 63:48)
  g1[2] = (rows << 16);    // tensor_dim1 lo16 (bits 95:80)
  g1[3] = (8 << 16);       // tile_dim0 = 8 (bits 127:112)
  g1[4] = rows;            // tile_dim1 (bits 143:128)
  g1[5] = 8;               // tensor_dim0_stride = 8 (bits 191:160)
  g1[6] = 0;
  g1[7] = 0;
  i32x4 z4 = {0, 0, 0, 0};
#if defined(__clang_major__) && (__clang_major__ >= 23)
  i32x8 z8 = {0, 0, 0, 0, 0, 0, 0, 0};
  __builtin_amdgcn_tensor_load_to_lds(g0, g1, z4, z4, z8, 0);
#else
  __builtin_amdgcn_tensor_load_to_lds(g0, g1, z4, z4, 0);
#endif
}
#endif  // HAVE_TDM

// ---------------------------------------------------------------------------
// Kernel 1: normalize codebook rows (9*1024 rows of 8) into d_ws.
// ---------------------------------------------------------------------------
__global__ void rvq_normcb_kernel(const float* __restrict__ cb,
                                  float* __restrict__ cbn) {
  int r = blockIdx.x * blockDim.x + threadIdx.x;
  if (r >= NCB * KC) return;
  const float* s = cb + (size_t)r * 8;
  float ss = 0.f;
#pragma unroll
  for (int j = 0; j < 8; ++j) ss += s[j] * s[j];
  float inv = 1.f / fmaxf(sqrtf(ss), 1e-12f);
  float* d = cbn + (size_t)r * 8;
#pragma unroll
  for (int j = 0; j < 8; ++j) d[j] = s[j] * inv;
}

// ---------------------------------------------------------------------------
// Kernel 2: fused RVQ. grid = 16 batches x 64 token-groups; block = 128
// threads = 4 waves; each wave owns 16 tokens.
// ---------------------------------------------------------------------------
__global__ __launch_bounds__(128, 1)
void rvq_fused_kernel(const float* __restrict__ z,
                      const float* __restrict__ w_in,
                      const float* __restrict__ b_in,
                      const float* __restrict__ w_out,
                      const float* __restrict__ b_out,
                      const float* __restrict__ cb,
                      const float* __restrict__ cbn,
                      float* __restrict__ out,
                      float* __restrict__ partials) {
  extern __shared__ float lds[];
  const int tid  = threadIdx.x;
  const int wave = tid >> 5;
  const int lane = tid & 31;
  const int h    = lane >> 4;     // lane half
  const int ln   = lane & 15;     // fragment row/col index
  const int k0   = h ? 2 : 0;     // K offset owned by this half (ISA layout)

  const int b  = blockIdx.x >> 6;
  const int tg = blockIdx.x & 63;
  const int t0 = tg * 64 + wave * 16;

  float* myres = lds + OFF_RES + wave * RES_PW;
  float* win   = lds + OFF_WIN;
  float* wout  = lds + OFF_WOUT;
  float* cbs   = lds + OFF_CBN;
  float* bouts = lds + OFF_BOUT;
  float* bins  = lds + OFF_BIN;
  float* enc_w = lds + OFF_ENC + wave * 160;

  const float* zb = z + (size_t)b * (DD * TT) + t0;

  // Zero w_in rows 8..15 once (columns 8..15 of the in_proj B operand stay
  // zero for the whole kernel -> no mask multiply in the K loop).
  for (int idx = tid; idx < 8 * 516; idx += 128) win[8 * 516 + idx] = 0.f;

  // Load residual tile, transposing [d][t] -> [tok][d] in LDS.
  for (int idx = lane; idx < DD * 4; idx += 32) {
    int d = idx >> 2, q = idx & 3;
    v4f v = *(const v4f*)(zb + (size_t)d * TT + 4 * q);
    myres[(4 * q + 0) * RES_STR + d] = v.x;
    myres[(4 * q + 1) * RES_STR + d] = v.y;
    myres[(4 * q + 2) * RES_STR + d] = v.z;
    myres[(4 * q + 3) * RES_STR + d] = v.w;
  }

  float loss_acc = 0.f;

  for (int i = 0; i < NCB; ++i) {
    __syncthreads();  // previous step's consumers done before restaging

    // ---- stage this codebook's weights into LDS ----
#if HAVE_TDM
    if (wave == 0) {  // one wave drives the Tensor Data Mover
      tdm_load_rows8_pad10(cbn + (size_t)i * KC * 8, OFF_CBN * 4u, KC);
      tdm_load_rows8_pad10(w_out + (size_t)i * DD * 8, OFF_WOUT * 4u, DD);
    }
#else
    {
      const float* wo = w_out + (size_t)i * DD * 8;
      for (int idx = tid; idx < DD * 4; idx += 128) {
        int d = idx >> 2, p = (idx & 3) * 2;
        *(v2f*)(wout + d * 10 + p) = *(const v2f*)(wo + d * 8 + p);
      }
      const float* cn = cbn + (size_t)i * KC * 8;
      for (int idx = tid; idx < KC * 4; idx += 128) {
        int c = idx >> 2, p = (idx & 3) * 2;
        *(v2f*)(cbs + c * 10 + p) = *(const v2f*)(cn + c * 8 + p);
      }
    }
#endif
    {
      const float* wi = w_in + (size_t)i * 8 * DD;
      for (int idx = tid; idx < 8 * DD; idx += 128) {
        int c = idx >> 9, d = idx & 511;
        win[c * 516 + d] = wi[idx];
      }
      for (int idx = tid; idx < DD; idx += 128)
        bouts[idx] = b_out[(size_t)i * DD + idx];
      if (tid < 8) bins[tid] = b_in[i * 8 + tid];
      // Warm L2 for next step's w_in while we compute this one.
      if (i + 1 < NCB && tid < 8)
        __builtin_prefetch(w_in + (size_t)(i + 1) * 8 * DD + (size_t)tid * DD,
                           0, 1);
    }
#if HAVE_TDM
    __builtin_amdgcn_s_wait_tensorcnt(0);
#endif
    __syncthreads();

    // ---- in_proj: enc(16 tok x 8 cdim) = res^T(16x512) @ w_in^T + b_in ----
    // M=tokens, N=cdims (cols 8..15 zero), K=512. Two interleaved
    // accumulators halve the WMMA RAW chain.
    v8f acc;
    {
      float binv = (ln < 8) ? bins[ln] : 0.f;
      v8f acc0, acc1;
#pragma unroll
      for (int j = 0; j < 8; ++j) { acc0[j] = binv; acc1[j] = 0.f; }
      const float* ap = myres + ln * RES_STR + k0;
      const float* bp = win + ln * 516 + k0;
#pragma unroll 2
      for (int kk = 0; kk < DD; kk += 8) {
        v2f a0 = *(const v2f*)(ap + kk);
        v2f b0 = *(const v2f*)(bp + kk);
        v2f a1 = *(const v2f*)(ap + kk + 4);
        v2f b1 = *(const v2f*)(bp + kk + 4);
        acc0 = wmma4(a0, b0, acc0);
        acc1 = wmma4(a1, b1, acc1);
      }
      acc = acc0 + acc1;
    }
    // Scatter enc to LDS [tok][cdim] and write latents (t-contiguous).
    if (ln < 8) {
#pragma unroll
      for (int j = 0; j < 8; ++j) enc_w[(j + 8 * h) * 10 + ln] = acc[j];
      float* lat = out + LAT_OFF +
                   ((size_t)b * (NCB * 8) + i * 8 + ln) * TT + t0 + 8 * h;
      v4f a0 = {acc[0], acc[1], acc[2], acc[3]};
      v4f a1 = {acc[4], acc[5], acc[6], acc[7]};
      *(v4f*)(lat)     = a0;
      *(v4f*)(lat + 4) = a1;
    }
    __syncthreads();

    // B fragments of enc (K=8 dims x N=16 tokens), reused by dist + loss.
    v2f eb0 = *(const v2f*)(enc_w + ln * 10 + k0);
    v2f eb1 = *(const v2f*)(enc_w + ln * 10 + k0 + 4);

    // ---- nearest code: argmax over cbn @ enc^T (M=codes, N=tokens) ----
    float best = -3.4e38f;
    int   bidx = 0;
    for (int nt = 0; nt < 64; ++nt) {
      const float* cr = cbs + (nt * 16 + ln) * 10 + k0;
      v2f a0 = *(const v2f*)(cr);
      v2f a1 = *(const v2f*)(cr + 4);
      v8f s = {0.f, 0.f, 0.f, 0.f, 0.f, 0.f, 0.f, 0.f};
      s = wmma4(a0, eb0, s);
      s = wmma4(a1, eb1, s);
#pragma unroll
      for (int j = 0; j < 8; ++j) {
        int cid = nt * 16 + 8 * h + j;   // ascending -> strict '>' keeps first
        float v = s[j];
        if (v > best) { best = v; bidx = cid; }
      }
    }
    {  // combine lane halves (each token lives in lanes ln and ln+16)
      float ov = __shfl_xor(best, 16, 32);
      int   oi = __shfl_xor(bidx, 16, 32);
      if (ov > best || (ov == best && oi < bidx)) { best = ov; bidx = oi; }
    }
    if (h == 0)
      out[CODES_OFF + ((size_t)b * NCB + i) * TT + t0 + ln] = (float)bidx;

    // ---- gather raw codebook vector for this token; losses ----
    const float* qv = cb + ((size_t)i * KC + bidx) * 8;
    v2f q0 = *(const v2f*)(qv + k0);
    v2f q1 = *(const v2f*)(qv + k0 + 4);
    {
      float d0 = eb0.x - q0.x, d1 = eb0.y - q0.y;
      float d2 = eb1.x - q1.x, d3 = eb1.y - q1.y;
      loss_acc += d0 * d0 + d1 * d1 + d2 * d2 + d3 * d3;
    }

    // ---- out_proj + residual update: res -= w_out @ z_q_i + b_out ----
    // M=512 dims in 32 tiles, K=8, N=16 tokens (B frags = q0,q1).
    for (int mt = 0; mt < 32; ++mt) {
      const float* wr = wout + (mt * 16 + ln) * 10 + k0;
      v2f a0 = *(const v2f*)(wr);
      v2f a1 = *(const v2f*)(wr + 4);
      int dbase = mt * 16 + 8 * h;
      v4f bo0 = *(const v4f*)(bouts + dbase);
      v4f bo1 = *(const v4f*)(bouts + dbase + 4);
      v8f o = {bo0.x, bo0.y, bo0.z, bo0.w, bo1.x, bo1.y, bo1.z, bo1.w};
      o = wmma4(a0, q0, o);
      o = wmma4(a1, q1, o);
      float* rp = myres + ln * RES_STR + dbase;
      v4f r0 = *(v4f*)(rp);
      v4f r1 = *(v4f*)(rp + 4);
      r0.x -= o[0]; r0.y -= o[1]; r0.z -= o[2]; r0.w -= o[3];
      r1.x -= o[4]; r1.y -= o[5]; r1.z -= o[6]; r1.w -= o[7];
      *(v4f*)(rp)     = r0;
      *(v4f*)(rp + 4) = r1;
    }
  }

  __syncthreads();  // all waves done with staged LDS before reuse below

  // ---- z_q = z - residual_final ----
  float* zq = out + ZQ_OFF + (size_t)b * (DD * TT) + t0;
  for (int idx = lane; idx < DD * 4; idx += 32) {
    int d = idx >> 2, q = idx & 3;
    v4f zv = *(const v4f*)(zb + (size_t)d * TT + 4 * q);
    v4f o;
    o.x = zv.x - myres[(4 * q + 0) * RES_STR + d];
    o.y = zv.y - myres[(4 * q + 1) * RES_STR + d];
    o.z = zv.z - myres[(4 * q + 2) * RES_STR + d];
    o.w = zv.w - myres[(4 * q + 3) * RES_STR + d];
    *(v4f*)(zq + (size_t)d * TT + 4 * q) = o;
  }

  // ---- deterministic loss reduction: wave -> block -> per-block partial ----
#pragma unroll
  for (int m = 16; m >= 1; m >>= 1) loss_acc += __shfl_xor(loss_acc, m, 32);
  if (lane == 0) lds[OFF_WIN + wave] = loss_acc;  // win area is dead now
  __syncthreads();
  if (tid == 0)
    partials[blockIdx.x] =
        lds[OFF_WIN] + lds[OFF_WIN + 1] + lds[OFF_WIN + 2] + lds[OFF_WIN + 3];
}

// ---------------------------------------------------------------------------
// Kernel 3: fixed-order final loss sum (cl == cbl numerically: same MSE).
// ---------------------------------------------------------------------------
__global__ void rvq_loss_kernel(const float* __restrict__ partials,
                                float* __restrict__ out) {
  if (threadIdx.x == 0 && blockIdx.x == 0) {
    float s = 0.f;
    for (int i = 0; i < NBLOCKS; ++i) s += partials[i];
    s *= (1.0f / (16.0f * 8.0f * 4096.0f));  // mean over (B, CDIM, T)
    out[CL_OFF]  = s;
    out[CBL_OFF] = s;
  }
}

// ---------------------------------------------------------------------------
extern "C" void kernel_launch(void* const* d_in, const int* in_sizes, int n_in,
                              void* d_out, int out_size, void* d_ws,
                              size_t ws_size, hipStream_t stream) {
  const float* z     = (const float*)d_in[0];
  const float* w_in  = (const float*)d_in[1];
  const float* b_in  = (const float*)d_in[2];
  const float* w_out = (const float*)d_in[3];
  const float* b_out = (const float*)d_in[4];
  const float* cb    = (const float*)d_in[5];
  float* out = (float*)d_out;
  float* cbn = (float*)d_ws;                       // 73728 floats
  float* partials = (float*)d_ws + WS_PART_OFF;    // 1024 floats

  rvq_normcb_kernel<<<dim3((NCB * KC + 255) / 256), dim3(256), 0, stream>>>(
      cb, cbn);

  size_t lds_bytes = (size_t)LDS_FLOATS * sizeof(float);  // ~226 KB (<320KB WGP)
  rvq_fused_kernel<<<dim3(NBLOCKS), dim3(128), lds_bytes, stream>>>(
      z, w_in, b_in, w_out, b_out, cb, cbn, out, partials);

  rvq_loss_kernel<<<dim3(1), dim3(32), 0, stream>>>(partials, out);
}